// VolTransformer_25202868093583
// MI455X (gfx1250) — compile-verified
//
#include <hip/hip_runtime.h>
#include <hip/hip_bf16.h>

typedef __attribute__((ext_vector_type(2))) float v2f;
typedef __attribute__((ext_vector_type(8))) float v8f;

#define WAVES_PER_BLOCK 8
#define TILES_PER_WAVE  4   // 4 x 16 = 64 points per wave per block pass

// out[b,h,n,p] = sum_c (pcd[b,h,c,p] + shift[h,c]) * R[h,c,n] * scales[h,n]
// One V_WMMA_F32_16X16X4_F32 processes 16 points:
//   A[m][k] = R[k][m]  (loop-invariant),  B[k][n] = x_k[point n],
//   D[m][n] = output component m of point n  (rows 0..2 of D used).
__global__ void __launch_bounds__(WAVES_PER_BLOCK * 32)
VolTransformer_25202868093583_kernel(const float* __restrict__ pcd,
                                     const float* __restrict__ logR,
                                     const float* __restrict__ shift,
                                     const float* __restrict__ scales,
                                     float* __restrict__ out,
                                     int H, long long P)
{
    const int bh = blockIdx.y;        // b*H + h
    const int h  = bh % H;

    // ---- Rodrigues: R = I + f1*K + f2*(w w^T - |w|^2 I), K = [w]x ----
    const float wx = logR[h * 3 + 0];
    const float wy = logR[h * 3 + 1];
    const float wz = logR[h * 3 + 2];
    const float t2  = wx * wx + wy * wy + wz * wz;
    const float t2c = fmaxf(t2, 1e-8f);
    const float th  = sqrtf(t2c);
    const float s   = __sinf(th);
    const float c   = __cosf(th);
    const float f1  = s / th;
    const float f2  = (1.0f - c) / t2c;

    // R[row=c][col=n]  (row = input component, col = output component)
    const float R00 = 1.0f + f2 * (wx * wx - t2);
    const float R01 = f1 * (-wz) + f2 * (wx * wy);
    const float R02 = f1 * ( wy) + f2 * (wx * wz);
    const float R10 = f1 * ( wz) + f2 * (wy * wx);
    const float R11 = 1.0f + f2 * (wy * wy - t2);
    const float R12 = f1 * (-wx) + f2 * (wy * wz);
    const float R20 = f1 * (-wy) + f2 * (wz * wx);
    const float R21 = f1 * ( wx) + f2 * (wz * wy);
    const float R22 = 1.0f + f2 * (wz * wz - t2);

    const int lane = threadIdx.x & 31;
    const int wave = threadIdx.x >> 5;
    const int m    = lane & 15;        // column (lower half) / row-within-half
    const bool lo  = lane < 16;

    // ---- A operand: A[m][k] = R[k][m], zero-padded. Layout:
    // a.x : lanes 0-15 -> A[m][0]=R[0][m] ; lanes 16-31 -> A[m][2]=R[2][m]
    // a.y : lanes 0-15 -> A[m][1]=R[1][m] ; lanes 16-31 -> A[m][3]=0
    const float Rr0 = (m == 0) ? R00 : (m == 1) ? R01 : (m == 2) ? R02 : 0.0f;
    const float Rr1 = (m == 0) ? R10 : (m == 1) ? R11 : (m == 2) ? R12 : 0.0f;
    const float Rr2 = (m == 0) ? R20 : (m == 1) ? R21 : (m == 2) ? R22 : 0.0f;
    v2f a;
    a.x = lo ? Rr0 : Rr2;
    a.y = lo ? Rr1 : 0.0f;

    const float sh0 = shift[h * 3 + 0];
    const float sh1 = shift[h * 3 + 1];
    const float sh2 = shift[h * 3 + 2];
    const float sc0 = scales[h * 3 + 0];
    const float sc1 = scales[h * 3 + 1];
    const float sc2 = scales[h * 3 + 2];

    const long long planeBase = (long long)bh * 3ll * P;
    const float* __restrict__ p0 = pcd + planeBase;      // component x plane
    const float* __restrict__ p1 = p0 + P;               // y
    const float* __restrict__ p2 = p1 + P;               // z
    float* __restrict__ o0 = out + planeBase;
    float* __restrict__ o1 = o0 + P;
    float* __restrict__ o2 = o1 + P;

    // lane-dependent source for B row-pair {K=0 | K=2}
    const float* __restrict__ pB0 = lo ? p0 : p2;
    const float shB0 = lo ? sh0 : sh2;

    const long long base =
        (long long)blockIdx.x * (WAVES_PER_BLOCK * 16 * TILES_PER_WAVE) +
        (long long)wave * (16 * TILES_PER_WAVE) + m;

    // ---- issue all loads up front (MLP), streaming / non-temporal ----
    v2f b[TILES_PER_WAVE];
#pragma unroll
    for (int t = 0; t < TILES_PER_WAVE; ++t) {
        const long long pt = base + (long long)t * 16;
        float v0 = 0.0f, v1 = 0.0f;
        if (pt < P) {
            v0 = __builtin_nontemporal_load(pB0 + pt);
            v1 = __builtin_nontemporal_load(p1 + pt);   // all lanes: same addrs as lower half, coalesces
        }
        b[t].x = v0 + shB0;
        b[t].y = lo ? (v1 + sh1) : 0.0f;                // K=3 row must be zero
    }

    // ---- matrix pipe: D = A x B (+0), then scale + streamed stores ----
#pragma unroll
    for (int t = 0; t < TILES_PER_WAVE; ++t) {
        v8f acc = {};
        acc = __builtin_amdgcn_wmma_f32_16x16x4_f32(
            /*neg_a=*/false, a, /*neg_b=*/false, b[t],
            /*c_mod=*/(short)0, acc, /*reuse_a=*/false, /*reuse_b=*/false);

        const long long pt = base + (long long)t * 16;
        if (lo && pt < P) {
            __builtin_nontemporal_store(acc[0] * sc0, o0 + pt);  // D row 0 = out comp 0
            __builtin_nontemporal_store(acc[1] * sc1, o1 + pt);  // D row 1 = out comp 1
            __builtin_nontemporal_store(acc[2] * sc2, o2 + pt);  // D row 2 = out comp 2
        }
    }
}

extern "C" void kernel_launch(void* const* d_in, const int* in_sizes, int n_in,
                              void* d_out, int out_size, void* d_ws, size_t ws_size,
                              hipStream_t stream) {
    const float* pcd    = (const float*)d_in[0];  // (B,H,3,P)
    const float* logR   = (const float*)d_in[1];  // (H,3)
    const float* shift  = (const float*)d_in[2];  // (H,3)
    const float* scales = (const float*)d_in[3];  // (H,3)
    float* out = (float*)d_out;

    const int H = in_sizes[1] / 3;                // 16
    const int B = 8;                              // per reference setup
    const long long P = (long long)in_sizes[0] / ((long long)B * H * 3);

    const long long ptsPerBlock = WAVES_PER_BLOCK * 16 * TILES_PER_WAVE;  // 512
    dim3 grid((unsigned)((P + ptsPerBlock - 1) / ptsPerBlock), (unsigned)(B * H));
    dim3 block(WAVES_PER_BLOCK * 32);

    VolTransformer_25202868093583_kernel<<<grid, block, 0, stream>>>(
        pcd, logR, shift, scales, out, H, P);
}